// SpanEmbeddings_53446573031784
// MI455X (gfx1250) — compile-verified
//
#include <hip/hip_runtime.h>
#include <hip/hip_bf16.h>

// ---------------------------------------------------------------------------
// Problem constants (from reference)
// ---------------------------------------------------------------------------
#define NUM_WORDS 8192
#define NUM_SPANS 8192
#define MAX_W     30
#define CTX_DIM   1024
#define HEAD_DIM  768
#define FEAT      256
#define SPAN_COLS (CTX_DIM + CTX_DIM + FEAT + HEAD_DIM)  // 3072

typedef __attribute__((ext_vector_type(2))) float v2f;
typedef __attribute__((ext_vector_type(4))) float v4f;   // native clang vector
typedef __attribute__((ext_vector_type(8))) float v8f;

// ---------------------------------------------------------------------------
// Kernel 1: head_scores = context_outputs @ W^T + b  via V_WMMA_F32_16X16X4_F32
//
// One wave handles 16 words (M=16). K=1024 swept in steps of 4.
// A (16x4 f32, 2 VGPR): lanes 0-15 hold M=lane, K=(0,1); lanes 16-31 hold
// M=lane-16, K=(2,3)  (per ISA 7.12.2 "32-bit A-Matrix 16x4").
// B (4x16): we broadcast W[k] into every column, so every column of D holds
// the same dot product -> column-layout details of B don't affect the result.
// D (16x16 f32, 8 VGPR): lane 0 VGPR j = (M=j, N=0); lane 16 VGPR j = (M=8+j).
// ---------------------------------------------------------------------------
__global__ __launch_bounds__(256) void head_scores_wmma_kernel(
    const float* __restrict__ ctx,   // [NUM_WORDS, CTX_DIM]
    const float* __restrict__ W,     // [CTX_DIM]
    const float* __restrict__ bias,  // [1]
    float* __restrict__ out)         // [NUM_WORDS]
{
    const int wave     = threadIdx.x >> 5;
    const int lane     = threadIdx.x & 31;
    const int wordBase = blockIdx.x * 128 + wave * 16;
    const int m        = lane & 15;
    const int kh       = (lane >> 4) << 1;  // 0 for lanes 0-15, 2 for 16-31

    const float* __restrict__ row = ctx + (size_t)(wordBase + m) * CTX_DIM + kh;
    const float* __restrict__ w2  = W + kh;

    v8f acc = {};
#pragma unroll 8
    for (int k = 0; k < CTX_DIM; k += 4) {
        v2f a;  a.x = row[k]; a.y = row[k + 1];   // contiguous -> b64 load
        v2f b;  b.x = w2[k];  b.y = w2[k + 1];    // broadcast W into all cols
        acc = __builtin_amdgcn_wmma_f32_16x16x4_f32(
            /*neg_a=*/false, a, /*neg_b=*/false, b,
            /*c_mod=*/(short)0, acc, /*reuse_a=*/false, /*reuse_b=*/false);
    }

    const float bv = bias[0];
    if (m == 0) {  // lane 0 -> words 0..7, lane 16 -> words 8..15 (N==0 column)
        const int base = wordBase + (lane >> 4) * 8;
#pragma unroll
        for (int j = 0; j < 8; ++j)
            __builtin_nontemporal_store(acc[j] + bv, out + base + j);
    }
}

// ---------------------------------------------------------------------------
// Kernel 2: one 192-thread block per span.
//   Phase A: indices + log-mask + masked softmax over <=30 scores (LDS).
//   Phase B: v4f copies of start/end context rows + width embedding.
//   Phase C: 30 gathered head_emb rows: NT-store to span_text_emb and
//            FMA-accumulate attention-weighted head embedding.
// head_emb (25 MB) + context (32 MB) stay hot in the 192 MB L2; all big
// output streams use non-temporal stores so they don't evict them.
// ---------------------------------------------------------------------------
__global__ __launch_bounds__(192) void span_gather_kernel(
    const float* __restrict__ head_emb,  // [NUM_WORDS, HEAD_DIM]
    const float* __restrict__ ctx,       // [NUM_WORDS, CTX_DIM]
    const int*   __restrict__ starts,    // [NUM_SPANS]
    const int*   __restrict__ ends,      // [NUM_SPANS]
    const float* __restrict__ wemb,      // [MAX_W, FEAT]
    const float* __restrict__ scores,    // [NUM_WORDS] (kernel 1 output)
    float* __restrict__ span_emb,        // [NUM_SPANS, SPAN_COLS]
    float* __restrict__ span_text,       // [NUM_SPANS, MAX_W, HEAD_DIM]
    float* __restrict__ idx_out,         // [NUM_SPANS, MAX_W] (as float)
    float* __restrict__ mask_out)        // [NUM_SPANS, MAX_W]
{
    const int s   = blockIdx.x;
    const int tid = threadIdx.x;

    __shared__ float attn[32];
    __shared__ int   sidx[32];

    const int start = starts[s];
    const int end   = ends[s];
    const int width = end - start + 1;
    const float NEG_INF = -__builtin_inff();

    if (tid < MAX_W) {
        int idx = start + tid;
        if (idx > NUM_WORDS - 1) idx = NUM_WORDS - 1;
        sidx[tid] = idx;
        const bool valid = tid < width;
        attn[tid] = valid ? scores[idx] : NEG_INF;
        __builtin_nontemporal_store((float)idx, idx_out + (size_t)s * MAX_W + tid);
        __builtin_nontemporal_store(valid ? 0.0f : NEG_INF,
                                    mask_out + (size_t)s * MAX_W + tid);
    }
    __syncthreads();

    if (tid == 0) {  // masked softmax over <=30 entries, serial (trivial)
        float mx = NEG_INF;
        for (int w = 0; w < width; ++w) mx = fmaxf(mx, attn[w]);
        float denom = 0.0f;
        for (int w = 0; w < width; ++w) {
            const float e = expf(attn[w] - mx);
            attn[w] = e;
            denom += e;
        }
        const float inv = 1.0f / denom;
        for (int w = 0; w < MAX_W; ++w)
            attn[w] = (w < width) ? attn[w] * inv : 0.0f;
    }
    __syncthreads();

    // ---- Phase B: span_emb = [start_row | end_row | width_emb | head_emb] --
    v4f*       ospan = (v4f*)(span_emb + (size_t)s * SPAN_COLS);
    const v4f* srow  = (const v4f*)(ctx + (size_t)start * CTX_DIM);
    const v4f* erow  = (const v4f*)(ctx + (size_t)end * CTX_DIM);
    const v4f* wrow  = (const v4f*)(wemb + (size_t)(width - 1) * FEAT);

    for (int i = tid; i < CTX_DIM / 4; i += 192) {          // 256 v4f each
        __builtin_nontemporal_store(srow[i], ospan + i);
        __builtin_nontemporal_store(erow[i], ospan + 256 + i);
    }
    if (tid < FEAT / 4)                                      // 64 v4f
        __builtin_nontemporal_store(wrow[tid], ospan + 512 + tid);

    // ---- Phase C: gather 30 head rows; each lane owns one v4f of 768 ------
    v4f  acc   = {0.0f, 0.0f, 0.0f, 0.0f};
    v4f* otext = (v4f*)(span_text + (size_t)s * MAX_W * HEAD_DIM);

    for (int w = 0; w < MAX_W; ++w) {
        const v4f* hrow = (const v4f*)(head_emb + (size_t)sidx[w] * HEAD_DIM);
        if (w + 1 < MAX_W)  // global_prefetch_b8 for the next gathered row
            __builtin_prefetch(((const v4f*)(head_emb +
                                (size_t)sidx[w + 1] * HEAD_DIM)) + tid, 0, 1);
        const v4f   v = hrow[tid];                  // L2-resident gather read
        const float a = attn[w];                    // 0 for invalid positions
        acc.x = fmaf(a, v.x, acc.x);
        acc.y = fmaf(a, v.y, acc.y);
        acc.z = fmaf(a, v.z, acc.z);
        acc.w = fmaf(a, v.w, acc.w);
        __builtin_nontemporal_store(v, otext + (size_t)w * (HEAD_DIM / 4) + tid);
    }
    __builtin_nontemporal_store(acc, ospan + 576 + tid);     // cols 2304..3071
}

// ---------------------------------------------------------------------------
// Launch. Output tuple layout (flat, in return order):
//   O1 span_emb        [8192, 3072]
//   O2 head_scores     [8192, 1]
//   O3 span_text_emb   [8192, 30, 768]
//   O4 span_indices    [8192, 30]   (written as float values)
//   O5 log_mask        [8192, 30]
// ---------------------------------------------------------------------------
extern "C" void kernel_launch(void* const* d_in, const int* in_sizes, int n_in,
                              void* d_out, int out_size, void* d_ws, size_t ws_size,
                              hipStream_t stream) {
    const float* head_emb = (const float*)d_in[0];
    const float* ctx      = (const float*)d_in[1];
    const int*   starts   = (const int*)d_in[2];
    const int*   ends     = (const int*)d_in[3];
    const float* wemb     = (const float*)d_in[4];
    const float* W        = (const float*)d_in[5];
    const float* bias     = (const float*)d_in[6];

    float* out      = (float*)d_out;
    float* o_span   = out;
    float* o_scores = o_span + (size_t)NUM_SPANS * SPAN_COLS;
    float* o_text   = o_scores + NUM_WORDS;
    float* o_idx    = o_text + (size_t)NUM_SPANS * MAX_W * HEAD_DIM;
    float* o_mask   = o_idx + (size_t)NUM_SPANS * MAX_W;

    head_scores_wmma_kernel<<<NUM_WORDS / 128, 256, 0, stream>>>(
        ctx, W, bias, o_scores);
    span_gather_kernel<<<NUM_SPANS, 192, 0, stream>>>(
        head_emb, ctx, starts, ends, wemb, o_scores,
        o_span, o_text, o_idx, o_mask);
}